// ResidualVectorQuantizer_79448305042050
// MI455X (gfx1250) — compile-verified
//
#include <hip/hip_runtime.h>
#include <hip/hip_bf16.h>

// Residual Vector Quantization for MI455X (gfx1250, wave32, WMMA).
//
// x:         (B=16, D=256, T=2048) f32
// codebooks: (L=8,  K=1024, D=256) f32
// out:       (B,D,T) f32  ++  (B,L,T) int32 indices (concatenated in d_out)
//
// One persistent kernel: each workgroup owns 32 consecutive t-rows of one
// batch, keeps the fp32 residual in LDS across all 8 layers, and computes
// the 32x1024 distance GEMM per layer with V_WMMA_F32_16X16X4_F32.
// output = x - final_residual (sum of quantized = x - residual).

typedef __attribute__((ext_vector_type(2))) float v2f;
typedef __attribute__((ext_vector_type(8))) float v8f;

constexpr int Bb = 16;
constexpr int Dd = 256;
constexpr int Tt = 2048;
constexpr int Ll = 8;
constexpr int Kk = 1024;

constexpr int ROWS = 32;       // t-rows per workgroup
constexpr int RSTR = 258;      // LDS row stride (even -> 8B-aligned b64, 2-way conflicts only)

// ---------------------------------------------------------------------------
// Prologue: wnorm[l*K + k] = ||codebooks[l][k][:]||^2   (8192 values, 32 KB ws)
// ---------------------------------------------------------------------------
__global__ __launch_bounds__(256) void rvq_wnorm_kernel(
    const float* __restrict__ cbs, float* __restrict__ wnorm) {
  const int lane = threadIdx.x & 31;
  const int wv   = threadIdx.x >> 5;
  const int code = blockIdx.x * 8 + wv;            // 1024 blocks * 8 waves = 8192 codes
  const float* p = cbs + (size_t)code * Dd;
  float s = 0.0f;
  for (int j = 0; j < Dd; j += 32) {               // coalesced: lane-contiguous
    float v = p[j + lane];
    s += v * v;
  }
  #pragma unroll
  for (int off = 16; off; off >>= 1) s += __shfl_xor(s, off, 32);
  if (lane == 0) wnorm[code] = s;
}

// ---------------------------------------------------------------------------
// Main persistent RVQ kernel
// ---------------------------------------------------------------------------
__global__ __launch_bounds__(256) void rvq_kernel(
    const float* __restrict__ x,
    const float* __restrict__ cbs,
    const float* __restrict__ wnorm,
    float* __restrict__ outF) {
  __shared__ float R[ROWS * RSTR];   // residual tile, f32
  __shared__ float argd[ROWS * 4];   // per-row best dist, per quarter-wave
  __shared__ int   argi[ROWS * 4];   // per-row best index, per quarter-wave
  __shared__ int   chosen[ROWS];

  const int tid    = threadIdx.x;
  const int lane   = tid & 31;
  const int wv     = tid >> 5;
  const int lane16 = lane & 15;
  const int h      = lane >> 4;      // lane half (C rows i vs i+8)
  const int mtile  = wv & 1;         // which 16-row M tile
  const int quart  = wv >> 1;        // which 256-code quarter of K
  const int mBase  = mtile * 16;

  const int blocksPerB = Tt / ROWS;  // 64
  const int b  = blockIdx.x / blocksPerB;
  const int t0 = (blockIdx.x % blocksPerB) * ROWS;

  // ---- load x tile into LDS residual (coalesced over t) -------------------
  const int jj = tid & 31;           // t offset within tile
  const int dg = tid >> 5;           // d group 0..7
  for (int d = dg; d < Dd; d += 8) {
    R[jj * RSTR + d] = x[(size_t)b * Dd * Tt + (size_t)d * Tt + t0 + jj];
  }
  __syncthreads();

  int* outI = (int*)(outF + (size_t)Bb * Dd * Tt);

  for (int l = 0; l < Ll; ++l) {
    const float* cb = cbs   + (size_t)l * Kk * Dd;
    const float* wn = wnorm + l * Kk;

    float bestd[8];
    int   besti[8];
    #pragma unroll
    for (int i = 0; i < 8; ++i) { bestd[i] = 3.4e38f; besti[i] = 0; }

    // ---- distance GEMM: this wave covers (16 rows) x (256 codes) ----------
    for (int nt = 0; nt < 16; ++nt) {
      const int n = quart * 256 + nt * 16 + lane16;   // global code id
      const float* bp = cb + (size_t)n * Dd;          // codebook row (L2-resident)
      v8f c = {0.f, 0.f, 0.f, 0.f, 0.f, 0.f, 0.f, 0.f};
      #pragma unroll 8
      for (int kk = 0; kk < Dd; kk += 4) {
        // A 16x4 f32 frag: lane(m,h) -> {R[m][kk+2h], R[m][kk+2h+1]} (ds_load_b64)
        v2f a = *(const v2f*)&R[(mBase + lane16) * RSTR + kk + 2 * h];
        // B 4x16 f32 frag: lane(n,h) -> {W[n][kk+2h], W[n][kk+2h+1]}
        v2f bb = *(const v2f*)(bp + kk + 2 * h);
        c = __builtin_amdgcn_wmma_f32_16x16x4_f32(
            false, a, false, bb, (short)0, c, false, false);
      }
      // dist' = ||w||^2 - 2 r.w  (||r||^2 dropped: constant per row)
      const float wnv = wn[n];
      #pragma unroll
      for (int i = 0; i < 8; ++i) {
        float dist = wnv - 2.0f * c[i];
        if (dist < bestd[i]) { bestd[i] = dist; besti[i] = n; }
      }
    }

    // ---- argmin reduce inside each 16-lane half (rows i / i+8) ------------
    #pragma unroll
    for (int i = 0; i < 8; ++i) {
      #pragma unroll
      for (int off = 8; off; off >>= 1) {
        float od = __shfl_xor(bestd[i], off, 32);
        int   oi = __shfl_xor(besti[i], off, 32);
        if (od < bestd[i] || (od == bestd[i] && oi < besti[i])) {
          bestd[i] = od; besti[i] = oi;
        }
      }
    }
    if (lane16 == 0) {            // lanes 0 and 16 hold the full half-reductions
      #pragma unroll
      for (int i = 0; i < 8; ++i) {
        int row = mBase + i + 8 * h;
        argd[row * 4 + quart] = bestd[i];
        argi[row * 4 + quart] = besti[i];
      }
    }
    __syncthreads();

    // ---- combine 4 quarter-wave candidates, emit indices ------------------
    if (tid < ROWS) {
      float bd = argd[tid * 4];
      int   bi = argi[tid * 4];
      #pragma unroll
      for (int q = 1; q < 4; ++q) {
        float dq = argd[tid * 4 + q];
        int   iq = argi[tid * 4 + q];
        if (dq < bd || (dq == bd && iq < bi)) { bd = dq; bi = iq; }
      }
      chosen[tid] = bi;
      outI[(size_t)b * Ll * Tt + (size_t)l * Tt + t0 + tid] = bi;
    }
    __syncthreads();

    // ---- residual -= selected codebook row (tid = d column, coalesced) ----
    for (int m = 0; m < ROWS; ++m) {
      R[m * RSTR + tid] -= cb[(size_t)chosen[m] * Dd + tid];
    }
    __syncthreads();
  }

  // ---- output = x - final residual (sum of quantized vectors) -------------
  for (int d = dg; d < Dd; d += 8) {
    size_t g = (size_t)b * Dd * Tt + (size_t)d * Tt + t0 + jj;
    outF[g] = x[g] - R[jj * RSTR + d];
  }
}

extern "C" void kernel_launch(void* const* d_in, const int* in_sizes, int n_in,
                              void* d_out, int out_size, void* d_ws, size_t ws_size,
                              hipStream_t stream) {
  const float* x   = (const float*)d_in[0];   // (B, D, T) f32
  const float* cbs = (const float*)d_in[1];   // (L, K, D) f32
  float* wnorm = (float*)d_ws;                // 8192 f32 = 32 KB scratch

  // 1) per-code squared norms
  rvq_wnorm_kernel<<<(Ll * Kk) / 8, 256, 0, stream>>>(cbs, wnorm);
  // 2) persistent RVQ: 16 batches * 64 t-tiles = 1024 workgroups
  rvq_kernel<<<Bb * (Tt / ROWS), 256, 0, stream>>>(x, cbs, wnorm, (float*)d_out);
}